// _TrainLevel_6889127543274
// MI455X (gfx1250) — compile-verified
//
#include <hip/hip_runtime.h>
#include <stdint.h>

#define NPB 256                     // points per block
#define BUCKET_MASK 0x3FFFFFu       // 2^22 - 1
#define P2 2654435761u
#define P3 805459861u

typedef float v4f __attribute__((ext_vector_type(4)));

__global__ __launch_bounds__(NPB) void hashgrid_trilerp_kernel(
    const float* __restrict__ pts,   // [N,3]
    const float* __restrict__ vox,   // [2^22, 4]
    float*       __restrict__ out,   // [N,4]
    int npts)
{
    __shared__ __align__(16) float spts[NPB * 3];   // 3 KB staging for this block's points

    const int tid        = threadIdx.x;
    const int block_base = blockIdx.x * NPB;        // first point index of this block

    // ---- Stage this block's points into LDS with async b128 copies (CDNA5 path) ----
    // 256 points * 3 floats = 768 floats = 192 float4 vectors; threads 0..191 move one each.
    if (tid < (NPB * 3) / 4) {
        const long long felem = (long long)block_base * 3 + (long long)tid * 4;
        const long long ftot  = (long long)npts * 3;
        if (felem + 4 <= ftot) {
            uint32_t lds_off = (uint32_t)(uintptr_t)(&spts[tid * 4]);
            uint64_t gaddr   = (uint64_t)(uintptr_t)(pts) + (uint64_t)felem * 4u;
            // copies 16 bytes from the global address in gaddr to the LDS address in lds_off, tracked by ASYNCcnt
            asm volatile("global_load_async_to_lds_b128 %0, %1, off"
                         :: "v"(lds_off), "v"(gaddr) : "memory");
        } else {
            // tail of the last block: guarded scalar copy
            #pragma unroll
            for (int k = 0; k < 4; ++k) {
                long long fe = felem + k;
                if (fe < ftot) spts[tid * 4 + k] = pts[fe];
            }
        }
    }
    asm volatile("s_wait_asynccnt 0x0" ::: "memory");
    __syncthreads();

    const int n = block_base + tid;
    if (n >= npts) return;

    const float px = spts[tid * 3 + 0];
    const float py = spts[tid * 3 + 1];
    const float pz = spts[tid * 3 + 2];

    // q = (p - smin) / res  (keep the true divide to match reference floor boundaries)
    const float qx = (px + 5.0f) / 0.05f;
    const float qy = (py + 5.0f) / 0.05f;
    const float qz = (pz + 5.0f) / 0.05f;
    const float bx = floorf(qx), by = floorf(qy), bz = floorf(qz);
    const float fx = qx - bx,   fy = qy - by,   fz = qz - bz;

    // Hash base voxel in uint32: mod 2^22 == low 22 bits of the uint32-wrapped sum.
    const uint32_t ibx = (uint32_t)(int)bx;
    const uint32_t iby = (uint32_t)(int)by;
    const uint32_t ibz = (uint32_t)(int)bz;
    const uint32_t h0  = ibx * 1u + iby * P2 + ibz * P3;

    // Issue all 8 gathers up front so the b128 loads overlap.
    const v4f* __restrict__ vf = (const v4f*)vox;
    v4f f[8];
    #pragma unroll
    for (int c = 0; c < 8; ++c) {
        uint32_t h = h0;
        if (c & 1) h += 1u;   // +x corner: prime 1
        if (c & 2) h += P2;   // +y corner
        if (c & 4) h += P3;   // +z corner
        f[c] = vf[h & BUCKET_MASK];
    }

    const float wx[2] = {1.0f - fx, fx};
    const float wy[2] = {1.0f - fy, fy};
    const float wz[2] = {1.0f - fz, fz};

    v4f acc = {0.f, 0.f, 0.f, 0.f};
    #pragma unroll
    for (int c = 0; c < 8; ++c) {
        const float w = wx[c & 1] * wy[(c >> 1) & 1] * wz[(c >> 2) & 1];
        acc.x = fmaf(w, f[c].x, acc.x);
        acc.y = fmaf(w, f[c].y, acc.y);
        acc.z = fmaf(w, f[c].z, acc.z);
        acc.w = fmaf(w, f[c].w, acc.w);
    }

    // Write-once output: nontemporal b128 store, don't pollute L2 (feature table lives there).
    __builtin_nontemporal_store(acc, (v4f*)out + n);
}

extern "C" void kernel_launch(void* const* d_in, const int* in_sizes, int n_in,
                              void* d_out, int out_size, void* d_ws, size_t ws_size,
                              hipStream_t stream) {
    const float* pts = (const float*)d_in[0];   // [N,3] float32
    const float* vox = (const float*)d_in[1];   // [2^22,4] float32
    // d_in[2] = primes (int64), d_in[3] = buckets (scalar) — baked into the kernel
    float* out = (float*)d_out;

    const int npts = in_sizes[0] / 3;
    const int blocks = (npts + NPB - 1) / NPB;
    hashgrid_trilerp_kernel<<<blocks, NPB, 0, stream>>>(pts, vox, out, npts);
}